// DGCNN_seg_25915832664277
// MI455X (gfx1250) — compile-verified
//
#include <hip/hip_runtime.h>
#include <hip/hip_bf16.h>
#include <stdint.h>

#define KNN   20
#define NEGS  0.2f
#define EPSB  1e-5f
#define NPTS  2048
#define NB    8

typedef __attribute__((ext_vector_type(2))) float v2f;
typedef __attribute__((ext_vector_type(8))) float v8f;

static __device__ __forceinline__ v8f wmma4(v2f a, v2f b, v8f c) {
  // V_WMMA_F32_16X16X4_F32 : exact fp32 16x16 tile MAC, K=4
  return __builtin_amdgcn_wmma_f32_16x16x4_f32(false, a, false, b, (short)0, c,
                                               false, false);
}

// Async global -> LDS copy (CDNA5 GLOBAL_LOAD_ASYNC_TO_LDS_B32, ASYNCcnt).
static __device__ __forceinline__ void async_g2l_b32(uint32_t lds_addr,
                                                     const float* gaddr) {
  asm volatile("global_load_async_to_lds_b32 %0, %1, off"
               :
               : "v"(lds_addr), "v"((unsigned long long)(uintptr_t)gaddr)
               : "memory");
}
static __device__ __forceinline__ void wait_async0() {
  asm volatile("s_wait_asynccnt 0x0" ::: "memory");
}

// ---------------------------------------------------------------- sqnorm
template <int C>
__global__ void sqnorm_kernel(const float* __restrict__ x, float* __restrict__ sq,
                              long bstride) {
  int i = blockIdx.x * blockDim.x + threadIdx.x;  // b*N + n
  if (i >= NB * NPTS) return;
  int b = i / NPTS, n = i % NPTS;
  const float* p = x + (long)b * bstride + n;
  float s = 0.f;
#pragma unroll 8
  for (int c = 0; c < C; ++c) {
    float t = p[(long)c * NPTS];
    s += t * t;
  }
  sq[i] = s;
}

// ---------------------------------------------------------------- knn top-k
// one wave32 per point row; per-lane register top-20 then wave merge.
template <int C>
__global__ void knn_kernel(const float* __restrict__ x, const float* __restrict__ sq,
                           int* __restrict__ idxg, long bstride) {
  __shared__ float cen[8][C];
  int tid = threadIdx.x;
  int wave = tid >> 5, lane = tid & 31;
  int row = blockIdx.x * 8 + wave;  // b*N + n
  int b = row / NPTS, n = row % NPTS;
  const float* xb = x + (long)b * bstride;
  for (int c = lane; c < C; c += 32) cen[wave][c] = xb[(long)c * NPTS + n];
  __syncthreads();
  float sqn = sq[row];

  float tv[KNN];
  int ti[KNN];
#pragma unroll
  for (int j = 0; j < KNN; ++j) { tv[j] = -3.0e38f; ti[j] = 0x7fffffff; }

  for (int m = lane; m < NPTS; m += 32) {
    float dot = 0.f;
#pragma unroll 16
    for (int c = 0; c < C; ++c) dot += xb[(long)c * NPTS + m] * cen[wave][c];
    float cv = 2.f * dot - sqn - sq[b * NPTS + m];  // = -||xi-xj||^2
    int ci = m;
#pragma unroll
    for (int j = 0; j < KNN; ++j) {  // branch-free sorted insert (best first)
      bool better = (cv > tv[j]) || (cv == tv[j] && ci < ti[j]);
      float nv = better ? cv : tv[j];
      int ni = better ? ci : ti[j];
      float ov = better ? tv[j] : cv;
      int oi = better ? ti[j] : ci;
      tv[j] = nv; ti[j] = ni; cv = ov; ci = oi;
    }
  }
  // merge across 32 lanes; tie-break: larger val, then lower index (jax top_k)
  for (int kk = 0; kk < KNN; ++kk) {
    float bvv = tv[0];
    int bii = ti[0];
#pragma unroll
    for (int off = 16; off; off >>= 1) {
      float ov = __shfl_xor(bvv, off, 32);
      int oi = __shfl_xor(bii, off, 32);
      if (ov > bvv || (ov == bvv && oi < bii)) { bvv = ov; bii = oi; }
    }
    if (ti[0] == bii) {  // unique owner pops its head
#pragma unroll
      for (int j = 0; j < KNN - 1; ++j) { tv[j] = tv[j + 1]; ti[j] = ti[j + 1]; }
      tv[KNN - 1] = -3.0e38f;
      ti[KNN - 1] = 0x7fffffff;
    }
    if (lane == 0) idxg[row * KNN + kk] = bii;
  }
}

// ---------------------------------------------------------------- edge conv
// block = 16-point tile, 8 waves over O-tiles. Per neighbor k: stage edge
// features in LDS, WMMA over 2C, fuse bias+BN+leaky, running max over k.
template <int C, int O>
__global__ void edgeconv_kernel(const float* __restrict__ xin, long bInStride,
                                const int* __restrict__ idxg,
                                const float* __restrict__ w,
                                const float* __restrict__ bias,
                                const float* __restrict__ g,
                                const float* __restrict__ be,
                                const float* __restrict__ bm,
                                const float* __restrict__ bv,
                                float* __restrict__ out, long bOutStride) {
  constexpr int TWOC = 2 * C;
  constexpr int KPAD = (TWOC + 3) & ~3;
  constexpr int KPADS = KPAD + 4;  // stride ≡ 4 (mod 64): conflict-free, 8B align
  constexpr int NOT = O / 16;      // O always multiple of 16 here
  constexpr int NT = (NOT + 7) / 8;  // o-tiles per wave
  constexpr bool PADW = (KPAD != TWOC);  // layer 1 only (2C=6 -> pad to 8)
  __shared__ int lidx[16 * KNN];
  __shared__ float As[16 * KPADS];
  __shared__ float wpad[PADW ? O * KPAD : 1];

  const int tilesN = NPTS / 16;
  int b = blockIdx.x / tilesN;
  int tile = blockIdx.x % tilesN;
  int tid = threadIdx.x;
  int wave = tid >> 5, lane = tid & 31;
  int hi = lane >> 4, col = lane & 15;
  int rowA = col;  // A-fragment row = lane%16
  const float* xb = xin + (long)b * bInStride;

  for (int i = tid; i < 16 * KNN; i += 256) {
    int r = i / KNN, kk = i % KNN;
    lidx[i] = idxg[((b * NPTS + tile * 16 + r) * KNN) + kk];
  }
  if constexpr (PADW) {  // zero-padded weight copy: unguarded b64 LDS reads later
    for (int i = tid; i < O * KPAD; i += 256) {
      int o = i / KPAD, c = i % KPAD;
      wpad[i] = (c < TWOC) ? w[o * TWOC + c] : 0.f;
    }
  }
  float bb[NT], sc[NT], mm[NT], eb[NT], macc[NT][8];
  bool valid[NT];
#pragma unroll
  for (int j = 0; j < NT; ++j) {
    int ot = wave + 8 * j;
    valid[j] = (ot < NOT);
    if (valid[j]) {
      int o = ot * 16 + col;
      bb[j] = bias[o];
      sc[j] = g[o] * rsqrtf(bv[o] + EPSB);
      mm[j] = bm[o];
      eb[j] = be[o];
    }
#pragma unroll
    for (int r = 0; r < 8; ++r) macc[j][r] = -3.0e38f;
  }
  __syncthreads();

  for (int k = 0; k < KNN; ++k) {
    for (int i = tid; i < 16 * KPAD; i += 256) {  // c-major fill, coalesced in r
      int r = i & 15, c = i >> 4;
      float val = 0.f;
      int n = tile * 16 + r;
      if (c < TWOC) {
        if (c < C) {
          int nb = lidx[r * KNN + k];
          val = xb[(long)c * NPTS + nb] - xb[(long)c * NPTS + n];
        } else {
          val = xb[(long)(c - C) * NPTS + n];
        }
      }
      As[r * KPADS + c] = val;
    }
    __syncthreads();
#pragma unroll
    for (int j = 0; j < NT; ++j) {
      if (valid[j]) {  // wave-uniform: EXEC stays all-ones for WMMA
        int ot = wave + 8 * j;
        int o = ot * 16 + col;
        const float* wrow = w + (long)o * TWOC;
        v8f acc = {0.f, 0.f, 0.f, 0.f, 0.f, 0.f, 0.f, 0.f};
#pragma unroll 8
        for (int kb = 0; kb < KPAD; kb += 4) {
          int c0 = kb + 2 * hi;
          v2f a = *(const v2f*)(As + rowA * KPADS + c0);
          v2f bf;
          if constexpr (PADW) {
            bf = *(const v2f*)(wpad + o * KPAD + c0);
          } else {  // layers 2-4: unguarded vector load
            bf = *(const v2f*)(wrow + c0);
          }
          acc = wmma4(a, bf, acc);
        }
#pragma unroll
        for (int r = 0; r < 8; ++r) {
          float h = acc[r] + bb[j];
          h = (h - mm[j]) * sc[j] + eb[j];
          h = (h >= 0.f) ? h : NEGS * h;
          macc[j][r] = fmaxf(macc[j][r], h);
        }
      }
    }
    __syncthreads();
  }
#pragma unroll
  for (int j = 0; j < NT; ++j) {
    if (valid[j]) {
      int ot = wave + 8 * j;
      int o = ot * 16 + col;
      float* ob = out + (long)b * bOutStride + (long)o * NPTS + tile * 16;
#pragma unroll
      for (int r = 0; r < 8; ++r) ob[r + 8 * hi] = macc[j][r];
    }
  }
}

// ---------------------------------------------------------------- pointwise GEMM
// Double-buffered A tile staged via async global->LDS (ASYNCcnt), overlapping
// the next chunk's memory traffic with the current chunk's WMMAs.
#define CHU 128
#define CHP (CHU + 4)
template <int C, int O, int FUSE>
__global__ void pointwise_kernel(const float* __restrict__ xin, long bInStride,
                                 const float* __restrict__ w,
                                 const float* __restrict__ bias,
                                 const float* __restrict__ g,
                                 const float* __restrict__ be,
                                 const float* __restrict__ bm,
                                 const float* __restrict__ bv,
                                 float* __restrict__ out, long bOutStride) {
  constexpr int NOT = (O + 15) / 16;
  constexpr int TPW = (NOT + 7) / 8;  // o-tiles per wave
  constexpr bool OGUARD = (O % 16) != 0;
  __shared__ float As[2][16 * CHP];
  const int tilesN = NPTS / 16;
  int b = blockIdx.x / tilesN;
  int tile = blockIdx.x % tilesN;
  int tid = threadIdx.x;
  int wave = tid >> 5, lane = tid & 31;
  int hi = lane >> 4, col = lane & 15;
  int rowA = col;
  const float* xb = xin + (long)b * bInStride + tile * 16;

  auto issue = [&](int cb, int p) {
    uint32_t base = (uint32_t)(uintptr_t)(&As[p][0]);
#pragma unroll 4
    for (int i = tid; i < 16 * CHU; i += 256) {
      int r = i & 15, c = i >> 4;  // coalesced over contiguous points
      async_g2l_b32(base + (uint32_t)(r * CHP + c) * 4u,
                    xb + (long)(cb + c) * NPTS + r);
    }
  };

  v8f acc[TPW];
#pragma unroll
  for (int j = 0; j < TPW; ++j)
    acc[j] = (v8f){0.f, 0.f, 0.f, 0.f, 0.f, 0.f, 0.f, 0.f};

  issue(0, 0);
  wait_async0();
  __syncthreads();

  int p = 0;
  for (int cb = 0; cb < C; cb += CHU) {
    if (cb + CHU < C) issue(cb + CHU, p ^ 1);  // prefetch next chunk
#pragma unroll
    for (int j = 0; j < TPW; ++j) {
      int ot = wave + 8 * j;
      if (ot < NOT) {  // wave-uniform guard (folds away unless O<128)
        int o = ot * 16 + col;
        bool ook = !OGUARD || (o < O);
        const float* wrow = w + (long)(ook ? o : 0) * C + cb;
#pragma unroll 8
        for (int kb = 0; kb < CHU; kb += 4) {
          int c0 = kb + 2 * hi;
          v2f a = *(const v2f*)(&As[p][rowA * CHP + c0]);
          v2f bf;
          if constexpr (!OGUARD) {
            bf = *(const v2f*)(wrow + c0);
          } else {
            bf.x = ook ? wrow[c0] : 0.f;
            bf.y = ook ? wrow[c0 + 1] : 0.f;
          }
          acc[j] = wmma4(a, bf, acc[j]);
        }
      }
    }
    wait_async0();   // own async loads for the next buffer are complete
    __syncthreads(); // all waves done reading As[p] and filling As[p^1]
    p ^= 1;
  }
#pragma unroll
  for (int j = 0; j < TPW; ++j) {
    int ot = wave + 8 * j;
    if (ot >= NOT) continue;
    int o = ot * 16 + col;
    if (!OGUARD || o < O) {
      float bbv = bias[o];
      float scv = 1.f, mmv = 0.f, ebv = 0.f;
      if constexpr (FUSE) {
        scv = g[o] * rsqrtf(bv[o] + EPSB);
        mmv = bm[o];
        ebv = be[o];
      }
      float* ob = out + (long)b * bOutStride + (long)o * NPTS + tile * 16;
#pragma unroll
      for (int r = 0; r < 8; ++r) {
        float h = acc[j][r] + bbv;
        if constexpr (FUSE) {
          h = (h - mmv) * scv + ebv;
          h = (h >= 0.f) ? h : NEGS * h;
        }
        ob[r + 8 * hi] = h;
      }
    }
  }
}

// ---------------------------------------------------------------- launch
extern "C" void kernel_launch(void* const* d_in, const int* in_sizes, int n_in,
                              void* d_out, int out_size, void* d_ws, size_t ws_size,
                              hipStream_t stream) {
  (void)in_sizes; (void)n_in; (void)out_size; (void)ws_size;
  const float* x = (const float*)d_in[0];
  const float *W[5], *Bs[5], *G[5], *BE[5], *M[5], *V[5];
  for (int i = 0; i < 5; ++i) {
    W[i]  = (const float*)d_in[1 + 6 * i + 0];
    Bs[i] = (const float*)d_in[1 + 6 * i + 1];
    G[i]  = (const float*)d_in[1 + 6 * i + 2];
    BE[i] = (const float*)d_in[1 + 6 * i + 3];
    M[i]  = (const float*)d_in[1 + 6 * i + 4];
    V[i]  = (const float*)d_in[1 + 6 * i + 5];
  }
  const float* w6 = (const float*)d_in[31];
  const float* b6 = (const float*)d_in[32];
  const float* w7 = (const float*)d_in[33];
  const float* b7 = (const float*)d_in[34];

  const long N = NPTS;
  float* xc = (float*)d_ws;                          // (B,512,N)
  float* xf = xc + (long)NB * 512 * N;               // (B,1024,N)
  float* x6 = xf + (long)NB * 1024 * N;              // (B,256,N)
  float* sq = x6 + (long)NB * 256 * N;               // (B,N)
  int* idx = (int*)(sq + (long)NB * N);              // (B,N,K)

  const int rows = NB * NPTS;
  dim3 blk(256);
  dim3 gRows((rows + 255) / 256);
  dim3 gKnn(rows / 8);
  dim3 gTile(NB * (NPTS / 16));

  // ---- EdgeConv 1: x(3) -> 64 @ xc[0:64]
  sqnorm_kernel<3><<<gRows, blk, 0, stream>>>(x, sq, 3L * N);
  knn_kernel<3><<<gKnn, blk, 0, stream>>>(x, sq, idx, 3L * N);
  edgeconv_kernel<3, 64><<<gTile, blk, 0, stream>>>(
      x, 3L * N, idx, W[0], Bs[0], G[0], BE[0], M[0], V[0], xc, 512L * N);
  // ---- EdgeConv 2: x1(64) -> 64 @ xc[64:128]
  sqnorm_kernel<64><<<gRows, blk, 0, stream>>>(xc, sq, 512L * N);
  knn_kernel<64><<<gKnn, blk, 0, stream>>>(xc, sq, idx, 512L * N);
  edgeconv_kernel<64, 64><<<gTile, blk, 0, stream>>>(
      xc, 512L * N, idx, W[1], Bs[1], G[1], BE[1], M[1], V[1], xc + 64L * N,
      512L * N);
  // ---- EdgeConv 3: x2(64) -> 128 @ xc[128:256]
  sqnorm_kernel<64><<<gRows, blk, 0, stream>>>(xc + 64L * N, sq, 512L * N);
  knn_kernel<64><<<gKnn, blk, 0, stream>>>(xc + 64L * N, sq, idx, 512L * N);
  edgeconv_kernel<64, 128><<<gTile, blk, 0, stream>>>(
      xc + 64L * N, 512L * N, idx, W[2], Bs[2], G[2], BE[2], M[2], V[2],
      xc + 128L * N, 512L * N);
  // ---- EdgeConv 4: x3(128) -> 256 @ xc[256:512]
  sqnorm_kernel<128><<<gRows, blk, 0, stream>>>(xc + 128L * N, sq, 512L * N);
  knn_kernel<128><<<gKnn, blk, 0, stream>>>(xc + 128L * N, sq, idx, 512L * N);
  edgeconv_kernel<128, 256><<<gTile, blk, 0, stream>>>(
      xc + 128L * N, 512L * N, idx, W[3], Bs[3], G[3], BE[3], M[3], V[3],
      xc + 256L * N, 512L * N);

  // conv5: 512 -> 1024, BN + leaky
  pointwise_kernel<512, 1024, 1><<<gTile, blk, 0, stream>>>(
      xc, 512L * N, W[4], Bs[4], G[4], BE[4], M[4], V[4], xf, 1024L * N);
  // conv6: 1024 -> 256, linear
  pointwise_kernel<1024, 256, 0><<<gTile, blk, 0, stream>>>(
      xf, 1024L * N, w6, b6, nullptr, nullptr, nullptr, nullptr, x6, 256L * N);
  // conv7: 256 -> 2, linear, into d_out (B,2,N)
  pointwise_kernel<256, 2, 0><<<gTile, blk, 0, stream>>>(
      x6, 256L * N, w7, b7, nullptr, nullptr, nullptr, nullptr, (float*)d_out,
      2L * N);
}